// NormalLoss_42623255445863
// MI455X (gfx1250) — compile-verified
//
#include <hip/hip_runtime.h>

// ---------------------------------------------------------------------------
// Bidirectional nearest-neighbor (normal loss) on gfx1250.
// S = P @ G^T computed 16x16-tile-at-a-time with V_WMMA_F32_16X16X4_F32.
// Row/col argmins tracked with packed sortable u64 keys (ds_min_u64 +
// global_atomic_min_u64).
// ---------------------------------------------------------------------------

typedef __attribute__((ext_vector_type(2))) float v2f;
typedef __attribute__((ext_vector_type(8))) float v8f;

#define WG_THREADS   256
#define ROWS_PER_WG  128          // 8 waves * 16 rows
#define COLS_PER_WG  2048
#define COL_TILES    (COLS_PER_WG / 16)

__device__ __forceinline__ unsigned long long packKey(float v, unsigned idx) {
    unsigned u = __float_as_uint(v);
    // map float -> monotonically increasing unsigned
    u = (u & 0x80000000u) ? ~u : (u | 0x80000000u);
    return ((unsigned long long)u << 32) | (unsigned long long)idx;
}

// ---------------------------------------------------------------------------
__global__ void nn_init(unsigned long long* __restrict__ rowk,
                        unsigned long long* __restrict__ colk,
                        int N, int M, float* __restrict__ out) {
    int i = blockIdx.x * blockDim.x + threadIdx.x;
    if (i == 0) out[0] = 0.0f;
    if (i < N) rowk[i] = ~0ull;
    if (i < M) colk[i] = ~0ull;
}

// ---------------------------------------------------------------------------
__global__ __launch_bounds__(WG_THREADS)
void nn_wmma(const float* __restrict__ p, const float* __restrict__ g,
             unsigned long long* __restrict__ rowk_g,
             unsigned long long* __restrict__ colk_g,
             int N, int M) {
    __shared__ float4             gtile[COLS_PER_WG];   // x,y,z,|g|^2 : 32 KB
    __shared__ unsigned long long colkey[COLS_PER_WG];  // 16 KB
    __shared__ unsigned long long rowkey[ROWS_PER_WG];  // 1 KB
    __shared__ float              psq_s[ROWS_PER_WG];   // 0.5 KB

    const int tid     = threadIdx.x;
    const int colBase = blockIdx.x * COLS_PER_WG;
    const int rowBase = blockIdx.y * ROWS_PER_WG;

    // Hint the upcoming streaming read of this block's G chunk (global_prefetch_b8).
    __builtin_prefetch(g + 3 * (size_t)colBase, 0, 1);

    // ---- stage G chunk to LDS (AoS float4), init local keys ----
    for (int c = tid; c < COLS_PER_WG; c += WG_THREADS) {
        const int j = colBase + c;
        float4 gv;
        if (j < M) {
            float x = g[3 * j + 0], y = g[3 * j + 1], z = g[3 * j + 2];
            gv = make_float4(x, y, z, x * x + y * y + z * z);
        } else {
            gv = make_float4(0.0f, 0.0f, 0.0f, 3.0e38f);  // never wins row-min
        }
        gtile[c]  = gv;
        colkey[c] = ~0ull;
    }
    for (int r = tid; r < ROWS_PER_WG; r += WG_THREADS) {
        const int i = rowBase + r;
        float q = 3.0e38f;                                // pad rows never win col-min
        if (i < N) {
            float x = p[3 * i + 0], y = p[3 * i + 1], z = p[3 * i + 2];
            q = x * x + y * y + z * z;
        }
        psq_s[r]  = q;
        rowkey[r] = ~0ull;
    }
    __syncthreads();

    const int wave   = tid >> 5;
    const int lane   = tid & 31;
    const int half   = lane >> 4;     // 0: lanes 0-15, 1: lanes 16-31
    const int lrow   = lane & 15;
    const int rstrip = rowBase + wave * 16;   // this wave's 16 rows

    // ---- A matrix (16x4 f32): lanes 0-15 hold K0/K1, lanes 16-31 hold K2/K3 ----
    v2f A;
    {
        int i = rstrip + lrow;
        i = (i < N) ? i : (N - 1);            // clamp; pad rows filtered at merge
        const float x = p[3 * i + 0];
        const float y = p[3 * i + 1];
        const float z = p[3 * i + 2];
        A.x = half ? z : x;                   // K=0 / K=2
        A.y = half ? 0.0f : y;                // K=1 / K=3 (K=3 padded with 0)
    }

    // |p|^2 for the 8 rows this lane covers in the C-matrix layout
    float psq[8];
#pragma unroll
    for (int r = 0; r < 8; ++r) psq[r] = psq_s[wave * 16 + half * 8 + r];

    // row-argmin accumulators (C-layout: VGPR r <-> row rstrip + half*8 + r)
    float    rmin[8];
    unsigned ridx[8];
#pragma unroll
    for (int r = 0; r < 8; ++r) { rmin[r] = 3.4e38f; ridx[r] = 0u; }

    // ---- stream column tiles ----
    for (int ct = 0; ct < COL_TILES; ++ct) {
        const int cbase = ct * 16;
        const float4 gv = gtile[cbase + lrow];

        // B matrix (4x16 f32), mirrored layout: lanes 0-15 K0/K1, lanes 16-31 K2/K3
        v2f B;
        B.x = half ? gv.z : gv.x;
        B.y = half ? 0.0f : gv.y;

        v8f S = {};
        S = __builtin_amdgcn_wmma_f32_16x16x4_f32(
                /*neg_a=*/false, A, /*neg_b=*/false, B,
                /*c_mod=*/(short)0, S, /*reuse_a=*/false, /*reuse_b=*/false);

        const float    gsq  = gv.w;
        const unsigned jcol = (unsigned)(colBase + cbase + lrow);

        float    cmin = 3.4e38f;
        unsigned crow = 0u;
#pragma unroll
        for (int r = 0; r < 8; ++r) {
            const float s  = S[r];                   // S[i,j] = p_i . g_j
            const float rm = fmaf(-2.0f, s, gsq);    // row ranking metric
            if (rm < rmin[r]) { rmin[r] = rm; ridx[r] = jcol; }
            const float cm = fmaf(-2.0f, s, psq[r]); // col ranking metric
            const unsigned irow = (unsigned)(rstrip + half * 8 + r);
            if (cm < cmin) { cmin = cm; crow = irow; }
        }
        // one ds_min_u64 per tile; lanes l and l+16 collide on the same column
        atomicMin(&colkey[cbase + lrow], packKey(cmin, crow));
    }

    // ---- fold row minima into LDS (16 lanes share each row slot) ----
#pragma unroll
    for (int r = 0; r < 8; ++r) {
        const int lr = wave * 16 + half * 8 + r;
        atomicMin(&rowkey[lr], packKey(rmin[r], ridx[r]));
    }
    __syncthreads();

    // ---- merge block-local winners into global key arrays ----
    for (int c = tid; c < COLS_PER_WG; c += WG_THREADS) {
        const int j = colBase + c;
        if (j < M) atomicMin(&colk_g[j], colkey[c]);
    }
    for (int r = tid; r < ROWS_PER_WG; r += WG_THREADS) {
        const int i = rowBase + r;
        if (i < N) atomicMin(&rowk_g[i], rowkey[r]);
    }
}

// ---------------------------------------------------------------------------
__global__ __launch_bounds__(WG_THREADS)
void nn_finalize(const float* __restrict__ pn, const float* __restrict__ gn,
                 const unsigned long long* __restrict__ rowk,
                 const unsigned long long* __restrict__ colk,
                 int N, int M, float* __restrict__ out) {
    __shared__ float red[WG_THREADS];
    const int i = blockIdx.x * blockDim.x + threadIdx.x;

    float acc = 0.0f;
    if (i < N) {
        const unsigned j = (unsigned)(rowk[i] & 0xffffffffull);
        const float d = pn[3 * i + 0] * gn[3 * j + 0]
                      + pn[3 * i + 1] * gn[3 * j + 1]
                      + pn[3 * i + 2] * gn[3 * j + 2];
        acc = (1.0f - d) / (float)N;
    } else if (i - N < M) {
        const int      j  = i - N;
        const unsigned ii = (unsigned)(colk[j] & 0xffffffffull);
        const float d = gn[3 * j + 0] * pn[3 * ii + 0]
                      + gn[3 * j + 1] * pn[3 * ii + 1]
                      + gn[3 * j + 2] * pn[3 * ii + 2];
        acc = (1.0f - d) / (float)M;
    }

    red[threadIdx.x] = acc;
    __syncthreads();
#pragma unroll
    for (int s = WG_THREADS / 2; s > 0; s >>= 1) {
        if (threadIdx.x < s) red[threadIdx.x] += red[threadIdx.x + s];
        __syncthreads();
    }
    if (threadIdx.x == 0) atomicAdd(out, red[0]);
}

// ---------------------------------------------------------------------------
extern "C" void kernel_launch(void* const* d_in, const int* in_sizes, int n_in,
                              void* d_out, int out_size, void* d_ws, size_t ws_size,
                              hipStream_t stream) {
    const float* pred_pts = (const float*)d_in[0];
    const float* pred_nrm = (const float*)d_in[1];
    const float* gt_pts   = (const float*)d_in[2];
    const float* gt_nrm   = (const float*)d_in[3];

    const int N = in_sizes[0] / 3;   // 8192
    const int M = in_sizes[2] / 3;   // 32768

    unsigned long long* rowk = (unsigned long long*)d_ws;        // N keys
    unsigned long long* colk = rowk + N;                         // M keys
    float* out = (float*)d_out;

    const int mx = (N > M) ? N : M;
    nn_init<<<(mx + WG_THREADS - 1) / WG_THREADS, WG_THREADS, 0, stream>>>(
        rowk, colk, N, M, out);

    dim3 grid((M + COLS_PER_WG - 1) / COLS_PER_WG,
              (N + ROWS_PER_WG - 1) / ROWS_PER_WG);
    nn_wmma<<<grid, WG_THREADS, 0, stream>>>(pred_pts, gt_pts, rowk, colk, N, M);

    const int tot = N + M;
    nn_finalize<<<(tot + WG_THREADS - 1) / WG_THREADS, WG_THREADS, 0, stream>>>(
        pred_nrm, gt_nrm, rowk, colk, N, M, out);
}